// DecoderRNN_24472723652950
// MI455X (gfx1250) — compile-verified
//
#include <hip/hip_runtime.h>
#include <hip/hip_bf16.h>
#include <math.h>

#define VV 50257
#define HH 1024
#define EE 512
#define LL 128

typedef __attribute__((ext_vector_type(2))) float v2f;
typedef __attribute__((ext_vector_type(8))) float v8f;

__device__ __forceinline__ float sigmoidf_(float x) {
    return 1.0f / (1.0f + expf(-x));
}

// ---------------------------------------------------------------------------
// Kernel 1: embed lookup, attention scores + softmax, attention context.
// Single block (tiny work: ~1.3 MB of reads total).
//   xa = [embed(512), h0(1024)]            (attn input, 1536)
//   xd = [embed(512), attn_out(1024)]      (a2d input, 1536)
// ---------------------------------------------------------------------------
__global__ void k1_prep(const int* __restrict__ tok,
                        const float* __restrict__ h0,
                        const float* __restrict__ enc,       // (L, H)
                        const float* __restrict__ emb,       // (V, E)
                        const float* __restrict__ attn_W,    // (L, E+H)
                        const float* __restrict__ attn_b,    // (L)
                        float* __restrict__ xa,
                        float* __restrict__ xd,
                        float* __restrict__ attnw_out) {
    __shared__ float s[LL];
    __shared__ float inv_sum;
    const int tid = threadIdx.x;
    const int t = tok[0];

    for (int i = tid; i < EE; i += 256) {
        float e = emb[(size_t)t * EE + i];
        xa[i] = e;
        xd[i] = e;
    }
    for (int i = tid; i < HH; i += 256) xa[EE + i] = h0[i];
    __syncthreads();

    if (tid < LL) {
        const float* wr = attn_W + (size_t)tid * (EE + HH);
        float acc = attn_b[tid];
        for (int k = 0; k < EE + HH; ++k) acc += wr[k] * xa[k];
        s[tid] = acc;
    }
    __syncthreads();

    if (tid == 0) {
        float mx = s[0];
        for (int l = 1; l < LL; ++l) mx = fmaxf(mx, s[l]);
        float sum = 0.f;
        for (int l = 0; l < LL; ++l) { float e = expf(s[l] - mx); s[l] = e; sum += e; }
        inv_sum = 1.0f / sum;
    }
    __syncthreads();

    if (tid < LL) {
        float w = s[tid] * inv_sum;
        s[tid] = w;
        attnw_out[tid] = w;
    }
    __syncthreads();

    // attn_out[h] = sum_l w[l] * enc[l][h]   (coalesced across threads)
    for (int hcol = tid; hcol < HH; hcol += 256) {
        float acc = 0.f;
        for (int l = 0; l < LL; ++l) acc += s[l] * enc[l * HH + hcol];
        xd[EE + hcol] = acc;
    }
}

// ---------------------------------------------------------------------------
// WMMA fp32 GEMV:  y = W1 @ x1 + (W2 @ x2) + b1 (+ b2), optional ReLU.
// One wave owns 16 rows; 8 waves / block -> 128 rows / block.
// A tile = 16x4 fp32 weight sub-block (global_load_b64 per lane),
// B tile = x broadcast into all 16 N columns (ds_load_b64 from LDS),
// accumulated with v_wmma_f32_16x16x4_f32.
// ---------------------------------------------------------------------------
__global__ void gemv2_wmma(const float* __restrict__ W1, const float* __restrict__ x1, int K1,
                           const float* __restrict__ W2, const float* __restrict__ x2, int K2,
                           const float* __restrict__ b1, const float* __restrict__ b2,
                           float* __restrict__ y, int R, int act) {
    __shared__ float xs[2048];
    const int tid = threadIdx.x;

    for (int i = tid; i < K1; i += 256) xs[i] = x1[i];
    if (W2 != nullptr) {
        for (int i = tid; i < K2; i += 256) xs[K1 + i] = x2[i];
    }
    __syncthreads();

    const int lane  = tid & 31;
    const int wave  = tid >> 5;
    const int row0  = blockIdx.x * 128 + wave * 16;
    if (row0 >= R) return;                       // wave-uniform exit

    const int m     = lane & 15;                 // M index held by this lane
    const int khalf = (lane >> 4) << 1;          // lanes 0-15: K{0,1}, 16-31: K{2,3}
    int rowA = row0 + m;
    if (rowA > R - 1) rowA = R - 1;              // clamp OOB loads (stores guarded)

    v8f c = {};                                  // C/D accumulator (rows row0..row0+15)

    const float* Ws[2]  = {W1, W2};
    const float* xsrc[2];
    int Ks[2]   = {K1, K2};
    int offs[2] = {0, K1};
    (void)xsrc;

    for (int p = 0; p < 2; ++p) {
        const float* W = Ws[p];
        const int K = Ks[p];
        if (W == nullptr || K <= 0) continue;
        const float* wrow = W + (size_t)rowA * K + khalf;
        const int off = offs[p] + khalf;

        int kb = 0;
        for (; kb + 64 <= K; kb += 64) {
            __builtin_prefetch(wrow + kb + 256, 0, 0);   // global_prefetch_b8, 1KB ahead
#pragma unroll
            for (int k0 = 0; k0 < 64; k0 += 4) {
                v2f a = *(const v2f*)(wrow + kb + k0);
                v2f b = *(const v2f*)(&xs[off + kb + k0]);
                c = __builtin_amdgcn_wmma_f32_16x16x4_f32(
                        false, a, false, b, (short)0, c, false, false);
            }
        }
        for (; kb < K; kb += 4) {
            v2f a = *(const v2f*)(wrow + kb);
            v2f b = *(const v2f*)(&xs[off + kb]);
            c = __builtin_amdgcn_wmma_f32_16x16x4_f32(
                    false, a, false, b, (short)0, c, false, false);
        }
    }

    // D layout: lane 0 holds rows row0..row0+7 in c[0..7]; lane 16 rows +8..+15.
    if ((lane & 15) == 0) {
        const int base = row0 + (lane >> 4) * 8;
#pragma unroll
        for (int j = 0; j < 8; ++j) {
            const int r = base + j;
            if (r < R) {
                float v = c[j];
                if (b1) v += b1[r];
                if (b2) v += b2[r];
                if (act) v = fmaxf(v, 0.f);
                y[r] = v;
            }
        }
    }
}

// ---------------------------------------------------------------------------
// LSTM pointwise: gate order i, f, g, o.
// ---------------------------------------------------------------------------
__global__ void k_lstm(const float* __restrict__ gates, const float* __restrict__ c0,
                       float* __restrict__ hnew, float* __restrict__ cnew) {
    const int i = blockIdx.x * blockDim.x + threadIdx.x;
    if (i < HH) {
        float ig = sigmoidf_(gates[0 * HH + i]);
        float fg = sigmoidf_(gates[1 * HH + i]);
        float gg = tanhf(gates[2 * HH + i]);
        float og = sigmoidf_(gates[3 * HH + i]);
        float c  = fg * c0[i] + ig * gg;
        float h  = og * tanhf(c);
        cnew[i] = c;
        hnew[i] = h;
    }
}

// ---------------------------------------------------------------------------
// log-sum-exp over logits (single block): red[0] = max + log(sum exp(x-max))
// ---------------------------------------------------------------------------
__global__ void k_logsumexp(const float* __restrict__ logits, int n, float* __restrict__ red) {
    __shared__ float buf[256];
    const int tid = threadIdx.x;
    float mx = -3.4e38f;
    for (int i = tid; i < n; i += 256) mx = fmaxf(mx, logits[i]);
    buf[tid] = mx;
    __syncthreads();
    for (int s = 128; s > 0; s >>= 1) {
        if (tid < s) buf[tid] = fmaxf(buf[tid], buf[tid + s]);
        __syncthreads();
    }
    mx = buf[0];
    __syncthreads();
    float sum = 0.f;
    for (int i = tid; i < n; i += 256) sum += expf(logits[i] - mx);
    buf[tid] = sum;
    __syncthreads();
    for (int s = 128; s > 0; s >>= 1) {
        if (tid < s) buf[tid] += buf[tid + s];
        __syncthreads();
    }
    if (tid == 0) red[0] = mx + logf(buf[0]);
}

__global__ void k_finalize(float* __restrict__ out, int n, const float* __restrict__ red) {
    const int i = blockIdx.x * blockDim.x + threadIdx.x;
    if (i < n) out[i] -= red[0];
}

// ---------------------------------------------------------------------------
extern "C" void kernel_launch(void* const* d_in, const int* in_sizes, int n_in,
                              void* d_out, int out_size, void* d_ws, size_t ws_size,
                              hipStream_t stream) {
    const int*   tok    = (const int*)d_in[0];
    const float* h0     = (const float*)d_in[1];
    const float* c0     = (const float*)d_in[2];
    const float* enc    = (const float*)d_in[3];
    const float* emb    = (const float*)d_in[4];
    const float* attn_W = (const float*)d_in[5];
    const float* attn_b = (const float*)d_in[6];
    const float* a2d_W  = (const float*)d_in[7];
    const float* a2d_b  = (const float*)d_in[8];
    const float* W_ih   = (const float*)d_in[9];
    const float* W_hh   = (const float*)d_in[10];
    const float* b_ih   = (const float*)d_in[11];
    const float* b_hh   = (const float*)d_in[12];
    const float* out_W  = (const float*)d_in[13];
    const float* out_b  = (const float*)d_in[14];

    float* out    = (float*)d_out;
    float* logits = out;                     // V  (finalized to log-probs in place)
    float* hnew   = out + VV;                // H
    float* cnew   = out + VV + HH;           // H
    float* attnw  = out + VV + 2 * HH;       // L

    float* ws    = (float*)d_ws;
    float* xa    = ws;                       // 1536: [embed, h0]
    float* xd    = ws + 1536;                // 1536: [embed, attn_out]
    float* dec   = ws + 3072;                // 1024: relu(a2d)
    float* gates = ws + 4096;                // 4096
    float* red   = ws + 8192;                // 1

    k1_prep<<<1, 256, 0, stream>>>(tok, h0, enc, emb, attn_W, attn_b, xa, xd, attnw);

    // dec = relu(a2d_W @ xd + a2d_b)                (1024 x 1536)
    gemv2_wmma<<<8, 256, 0, stream>>>(a2d_W, xd, EE + HH,
                                      nullptr, nullptr, 0,
                                      a2d_b, nullptr, dec, HH, 1);

    // gates = W_ih @ dec + b_ih + W_hh @ h0 + b_hh  (4096 x 1024, x2)
    gemv2_wmma<<<32, 256, 0, stream>>>(W_ih, dec, HH,
                                       W_hh, h0, HH,
                                       b_ih, b_hh, gates, 4 * HH, 0);

    k_lstm<<<4, 256, 0, stream>>>(gates, c0, hnew, cnew);

    // logits = out_W @ h_new + out_b                (50257 x 1024, 206 MB -> BW bound)
    gemv2_wmma<<<(VV + 127) / 128, 256, 0, stream>>>(out_W, hnew, HH,
                                                     nullptr, nullptr, 0,
                                                     out_b, nullptr, logits, VV, 0);

    k_logsumexp<<<1, 256, 0, stream>>>(logits, VV, red);
    k_finalize<<<(VV + 255) / 256, 256, 0, stream>>>(logits, VV, red);
}